// AttentionLayer_28389733826637
// MI455X (gfx1250) — compile-verified
//
#include <hip/hip_runtime.h>
#include <math.h>

// Problem constants (B,S,D) = (4, 2048, 1024)
#define BATCH   4
#define SEQ     2048
#define DMODEL  1024
#define ROWS    (BATCH * SEQ)          // 8192 total query rows
#define SCALE   32.0f                  // sqrt(DMODEL): reference MULTIPLIES by sqrt(d)

typedef float v2f __attribute__((ext_vector_type(2)));
typedef float v8f __attribute__((ext_vector_type(8)));

// D = A(16x4, f32) * B(4x16, f32) + C(16x16, f32), wave32.
// A layout (ISA 7.12.2): lane l<16 -> row M=l, VGPR0/1 = K0/K1; lanes 16..31 -> K2/K3.
// B layout (row-striped): VGPR0 = B[K=0+2*half][N=lane%16], VGPR1 = B[K=1+2*half][...].
// C/D: VGPR r -> element (M = r + 8*half, N = lane%16).
__device__ __forceinline__ v8f wmma_f32(v2f a, v2f b, v8f c) {
  return __builtin_amdgcn_wmma_f32_16x16x4_f32(
      /*neg_a=*/false, a, /*neg_b=*/false, b,
      /*c_mod=*/(short)0, c, /*reuse_a=*/false, /*reuse_b=*/false);
}

// ---------------------------------------------------------------------------
// K1: Q = X @ WQK^T, V = X @ WOV^T.
// One wave -> 32x32 tile of BOTH Q and V: 2 A-frags (X) x 2 B-frags per weight.
// Per k-step: 6 b64 loads -> 8 WMMA.
// Tiles: (ROWS/32)=256 x (DMODEL/32)=32 = 8192 waves.
// ---------------------------------------------------------------------------
__global__ __launch_bounds__(256) void qv_gemm(const float* __restrict__ X,
                                               const float* __restrict__ WQK,
                                               const float* __restrict__ WOV,
                                               float* __restrict__ Q,
                                               float* __restrict__ V) {
  const int wave = blockIdx.x * (blockDim.x >> 5) + (threadIdx.x >> 5);
  const int lane = threadIdx.x & 31;
  const int half = lane >> 4;
  const int l    = lane & 15;
  const int NT2 = DMODEL / 32;               // 32
  const int mt = wave / NT2;                 // 0..255
  const int nt = wave % NT2;                 // 0..31
  const int m0 = mt * 32, n0 = nt * 32;

  const float* a0 = X   + (size_t)(m0 + l) * DMODEL + 2 * half;
  const float* a1 = a0 + (size_t)16 * DMODEL;
  const float* q0 = WQK + (size_t)(n0 + l) * DMODEL + 2 * half;   // B[k][n]=W[n][k]
  const float* q1 = q0 + (size_t)16 * DMODEL;
  const float* v0 = WOV + (size_t)(n0 + l) * DMODEL + 2 * half;
  const float* v1 = v0 + (size_t)16 * DMODEL;

  v8f cq00 = {}, cq01 = {}, cq10 = {}, cq11 = {};
  v8f cv00 = {}, cv01 = {}, cv10 = {}, cv11 = {};
#pragma unroll 2
  for (int k = 0; k < DMODEL; k += 4) {
    v2f fa0 = *(const v2f*)(a0 + k);
    v2f fa1 = *(const v2f*)(a1 + k);
    v2f fq0 = *(const v2f*)(q0 + k);
    v2f fq1 = *(const v2f*)(q1 + k);
    v2f fv0 = *(const v2f*)(v0 + k);
    v2f fv1 = *(const v2f*)(v1 + k);
    cq00 = wmma_f32(fa0, fq0, cq00);
    cq01 = wmma_f32(fa0, fq1, cq01);
    cq10 = wmma_f32(fa1, fq0, cq10);
    cq11 = wmma_f32(fa1, fq1, cq11);
    cv00 = wmma_f32(fa0, fv0, cv00);
    cv01 = wmma_f32(fa0, fv1, cv01);
    cv10 = wmma_f32(fa1, fv0, cv10);
    cv11 = wmma_f32(fa1, fv1, cv11);
  }
#pragma unroll
  for (int r = 0; r < 8; ++r) {
    const size_t r0 = (size_t)(m0 + r + 8 * half);
    const size_t r1 = r0 + 16;
    Q[r0 * DMODEL + n0 + l]      = cq00[r];
    Q[r0 * DMODEL + n0 + 16 + l] = cq01[r];
    Q[r1 * DMODEL + n0 + l]      = cq10[r];
    Q[r1 * DMODEL + n0 + 16 + l] = cq11[r];
    V[r0 * DMODEL + n0 + l]      = cv00[r];
    V[r0 * DMODEL + n0 + 16 + l] = cv01[r];
    V[r1 * DMODEL + n0 + l]      = cv10[r];
    V[r1 * DMODEL + n0 + 16 + l] = cv11[r];
  }
}

// ---------------------------------------------------------------------------
// K2: raw scores Sc = (Q @ X^T) * 32, causal-masked, into the A region of
// d_out. One wave -> 32x64 supertile: 2 A-frags (Q) x 4 B-frags (X),
// 6 b64 loads -> 8 WMMA per k-step (~10.7 f32 FLOP per VMEM byte per lane).
// Supertiles fully above the diagonal skipped wave-uniformly (EXEC all-ones
// for every executed WMMA). Waves: BATCH * 64 * 32 = 8192.
// ---------------------------------------------------------------------------
__global__ __launch_bounds__(256) void score_gemm(const float* __restrict__ Q,
                                                  const float* __restrict__ X,
                                                  float* __restrict__ Araw) {
  const int wave = blockIdx.x * (blockDim.x >> 5) + (threadIdx.x >> 5);
  const int lane = threadIdx.x & 31;
  const int half = lane >> 4;
  const int l    = lane & 15;
  const int MT = SEQ / 32;   // 64
  const int TT = SEQ / 64;   // 32
  const int b  = wave / (MT * TT);
  int rem      = wave % (MT * TT);
  const int mt = rem / TT;
  const int tt = rem % TT;
  const int m0 = mt * 32, t0 = tt * 64;
  if (t0 > m0 + 31) return;  // fully above causal diagonal (uniform per wave)

  const float* qa0 = Q + (size_t)(b * SEQ + m0 + l) * DMODEL + 2 * half;
  const float* qa1 = qa0 + (size_t)16 * DMODEL;
  const float* xb0 = X + (size_t)(b * SEQ + t0 + l) * DMODEL + 2 * half;
  const float* xb1 = xb0 + (size_t)16 * DMODEL;
  const float* xb2 = xb0 + (size_t)32 * DMODEL;
  const float* xb3 = xb0 + (size_t)48 * DMODEL;

  v8f c00 = {}, c01 = {}, c02 = {}, c03 = {};
  v8f c10 = {}, c11 = {}, c12 = {}, c13 = {};
#pragma unroll 2
  for (int k = 0; k < DMODEL; k += 4) {
    v2f fa0 = *(const v2f*)(qa0 + k);
    v2f fa1 = *(const v2f*)(qa1 + k);
    v2f fb0 = *(const v2f*)(xb0 + k);
    v2f fb1 = *(const v2f*)(xb1 + k);
    v2f fb2 = *(const v2f*)(xb2 + k);
    v2f fb3 = *(const v2f*)(xb3 + k);
    c00 = wmma_f32(fa0, fb0, c00);
    c01 = wmma_f32(fa0, fb1, c01);
    c02 = wmma_f32(fa0, fb2, c02);
    c03 = wmma_f32(fa0, fb3, c03);
    c10 = wmma_f32(fa1, fb0, c10);
    c11 = wmma_f32(fa1, fb1, c11);
    c12 = wmma_f32(fa1, fb2, c12);
    c13 = wmma_f32(fa1, fb3, c13);
  }

  float* out = Araw + (size_t)b * SEQ * SEQ;
  v8f* acc[2][4] = {{&c00, &c01, &c02, &c03}, {&c10, &c11, &c12, &c13}};
#pragma unroll
  for (int i = 0; i < 2; ++i) {
#pragma unroll
    for (int j = 0; j < 4; ++j) {
      const v8f& c = *acc[i][j];
#pragma unroll
      for (int r = 0; r < 8; ++r) {
        const int m = m0 + i * 16 + r + 8 * half;
        const int t = t0 + j * 16 + l;
        float s = c[r] * SCALE;
        if (t > m) s = -3.0e38f;       // masked; softmax pass zeros these anyway
        out[(size_t)m * SEQ + t] = s;
      }
    }
  }
}

// ---------------------------------------------------------------------------
// K3: in-place row softmax over the causal prefix; zero above the diagonal
// (also covers supertiles K2 never wrote). One 256-thread block per row.
// ---------------------------------------------------------------------------
__global__ __launch_bounds__(256) void softmax_rows(float* __restrict__ A) {
  __shared__ float red[256];
  const int row = blockIdx.x;          // 0 .. ROWS-1
  const int m   = row % SEQ;
  const int tid = threadIdx.x;
  float* p = A + (size_t)row * SEQ;
  const int valid = m + 1;

  float lmax = -3.4e38f;
  for (int t = tid; t < valid; t += 256) lmax = fmaxf(lmax, p[t]);
  red[tid] = lmax; __syncthreads();
  for (int s = 128; s > 0; s >>= 1) {
    if (tid < s) red[tid] = fmaxf(red[tid], red[tid + s]);
    __syncthreads();
  }
  const float rmax = red[0]; __syncthreads();

  float lsum = 0.0f;
  for (int t = tid; t < valid; t += 256) lsum += expf(p[t] - rmax);
  red[tid] = lsum; __syncthreads();
  for (int s = 128; s > 0; s >>= 1) {
    if (tid < s) red[tid] += red[tid + s];
    __syncthreads();
  }
  const float rinv = 1.0f / red[0];

  for (int t = tid; t < SEQ; t += 256)
    p[t] = (t < valid) ? expf(p[t] - rmax) * rinv : 0.0f;
}

// ---------------------------------------------------------------------------
// K4: Y = A @ V with causal k-truncation, fused residual: out = X + Y.
// One wave -> 32x32 tile: 2 A-frags x 2 B-frags -> 4 WMMA from
// 2 b64 + 4 b32 loads per k-step. Waves: BATCH * 64 * 32 = 8192.
// ---------------------------------------------------------------------------
__global__ __launch_bounds__(256) void out_gemm(const float* __restrict__ A,
                                                const float* __restrict__ V,
                                                const float* __restrict__ X,
                                                float* __restrict__ Y) {
  const int wave = blockIdx.x * (blockDim.x >> 5) + (threadIdx.x >> 5);
  const int lane = threadIdx.x & 31;
  const int half = lane >> 4;
  const int l    = lane & 15;
  const int MT = SEQ / 32;      // 64
  const int NT = DMODEL / 32;   // 32
  const int b  = wave / (MT * NT);
  int rem      = wave % (MT * NT);
  const int mt = rem / NT, nt = rem % NT;
  const int m0 = mt * 32, n0 = nt * 32;

  const float* aa0 = A + (size_t)(b * SEQ + m0 + l) * SEQ + 2 * half;
  const float* aa1 = aa0 + (size_t)16 * SEQ;
  const float* vb  = V + (size_t)(b * SEQ) * DMODEL;

  const int kend = m0 + 32;     // A[m][t] == 0 for t > m, so truncate k-loop
  v8f c00 = {}, c01 = {}, c10 = {}, c11 = {};
#pragma unroll 4
  for (int k = 0; k < kend; k += 4) {
    v2f fa0 = *(const v2f*)(aa0 + k);
    v2f fa1 = *(const v2f*)(aa1 + k);
    const size_t row0 = (size_t)(k + 2 * half) * DMODEL;
    const size_t row1 = row0 + DMODEL;
    v2f fb0, fb1;
    fb0.x = vb[row0 + n0 + l];
    fb0.y = vb[row1 + n0 + l];
    fb1.x = vb[row0 + n0 + 16 + l];
    fb1.y = vb[row1 + n0 + 16 + l];
    c00 = wmma_f32(fa0, fb0, c00);
    c01 = wmma_f32(fa0, fb1, c01);
    c10 = wmma_f32(fa1, fb0, c10);
    c11 = wmma_f32(fa1, fb1, c11);
  }
#pragma unroll
  for (int r = 0; r < 8; ++r) {
    const size_t g0   = (size_t)(b * SEQ + m0 + r + 8 * half);
    const size_t g1   = g0 + 16;
    const size_t i00  = g0 * DMODEL + n0 + l;
    const size_t i01  = g0 * DMODEL + n0 + 16 + l;
    const size_t i10  = g1 * DMODEL + n0 + l;
    const size_t i11  = g1 * DMODEL + n0 + 16 + l;
    Y[i00] = X[i00] + c00[r];
    Y[i01] = X[i01] + c01[r];
    Y[i10] = X[i10] + c10[r];
    Y[i11] = X[i11] + c11[r];
  }
}

// ---------------------------------------------------------------------------
// Host launcher. Inputs: X (f32 B,S,D), mask (bool, ignored: causal is static),
// WQK (f32 D,D), WOV (f32 D,D). Output: [X+Y (B,S,D)] ++ [A (B,S,S)], f32.
// Workspace: Q (32 MB) + V (32 MB) in d_ws.
// ---------------------------------------------------------------------------
extern "C" void kernel_launch(void* const* d_in, const int* in_sizes, int n_in,
                              void* d_out, int out_size, void* d_ws, size_t ws_size,
                              hipStream_t stream) {
  (void)in_sizes; (void)n_in; (void)out_size; (void)ws_size;
  const float* X   = (const float*)d_in[0];
  const float* WQK = (const float*)d_in[2];
  const float* WOV = (const float*)d_in[3];

  float* Yout = (float*)d_out;                          // B*S*D residual output
  float* Aout = Yout + (size_t)ROWS * DMODEL;           // B*S*S attention matrix

  float* Q = (float*)d_ws;                              // 8192 x 1024 f32
  float* V = Q + (size_t)ROWS * DMODEL;                 // 8192 x 1024 f32

  // 8 waves per 256-thread block; 8192 waves per GEMM kernel.
  qv_gemm     <<<1024, 256, 0, stream>>>(X, WQK, WOV, Q, V);
  score_gemm  <<<1024, 256, 0, stream>>>(Q, X, Aout);
  softmax_rows<<<ROWS, 256, 0, stream>>>(Aout);
  out_gemm    <<<1024, 256, 0, stream>>>(Aout, V, X, Yout);
}